// GraphAttentionLayer_58832462020771
// MI455X (gfx1250) — compile-verified
//
#include <hip/hip_runtime.h>

#define N_TOT  8000
#define MM     32
#define F_IN   64
#define F_OUT  32
#define NHEAD  4

typedef float v2f __attribute__((ext_vector_type(2)));
typedef float v8f __attribute__((ext_vector_type(8)));

#define HOT_STRIDE 68   // 256-row h_other tile, padded (4*r+k banks -> conflict-free lanes)
#define G_STRIDE   68   // g rows padded likewise for WMMA A-fragment reads

// LDS layout (floats):
//   hot  : [256][68]      @ 0      (17408)
//   g    : [64][68]       @ 17408  (4352)    rows = h*16 + n_local
//   alpha: [4][16][32]    @ 21760  (2048)
//   uv   : [2][4][64]     @ 23808  (512)
#define SM_HOT   0
#define SM_G     17408
#define SM_ALPHA 21760
#define SM_UV    23808
#define SM_FLOATS 24320
#define SM_BYTES  (SM_FLOATS * 4)

// --- prep: u[h][k] = sum_o W[h][k][o]*a[o], v[h][k] = sum_o W[h][k][o]*a[32+o]
__global__ void __launch_bounds__(256) gat_prep(const float* __restrict__ w,
                                                const float* __restrict__ a,
                                                float* __restrict__ uv) {
  const int t = threadIdx.x;        // 0..255 -> h = t>>6, k = t&63
  const int h = t >> 6, k = t & 63;
  const float* wr = w + (h * F_IN + k) * F_OUT;
  float su = 0.f, sv = 0.f;
#pragma unroll
  for (int o = 0; o < F_OUT; ++o) {
    const float ww = wr[o];
    su += ww * a[o];
    sv += ww * a[F_OUT + o];
  }
  uv[t]       = su;
  uv[256 + t] = sv;
}

__global__ void __launch_bounds__(256) gat_main(const float* __restrict__ hs,
                                                const float* __restrict__ ho,
                                                const float* __restrict__ w,
                                                const float* __restrict__ bias,
                                                const float* __restrict__ uv_g,
                                                float* __restrict__ out) {
  extern __shared__ float sm[];
  float* hot = sm + SM_HOT;
  float* g   = sm + SM_G;
  float* alp = sm + SM_ALPHA;
  float* uv  = sm + SM_UV;

  const int t    = threadIdx.x;
  const int lane = t & 31;
  const int wave = t >> 5;                 // 8 waves; wave == local n (per pass)
  const int n0   = blockIdx.x * 16;        // 16 n-values per block

  uv[t]       = uv_g[t];
  uv[t + 256] = uv_g[t + 256];
  __syncthreads();

  for (int p = 0; p < 2; ++p) {
    const int  nloc = p * 8 + wave;                         // n within chunk
    const long row  = (long)(n0 + p * 8) * MM + t;          // global (n,m) row
    const float* hsr = hs + row * F_IN;
    const float* hor = ho + row * F_IN;

    // ---- stage 1: stream rows, score dots, stage h_other into LDS ----
    float su[NHEAD] = {0.f, 0.f, 0.f, 0.f};
    float sv[NHEAD] = {0.f, 0.f, 0.f, 0.f};
#pragma unroll
    for (int c = 0; c < F_IN / 4; ++c) {
      const float4 s4 = ((const float4*)hsr)[c];
      const float4 o4 = ((const float4*)hor)[c];
      ((float4*)&hot[t * HOT_STRIDE])[c] = o4;
#pragma unroll
      for (int h = 0; h < NHEAD; ++h) {
        const float4 u4 = ((const float4*)&uv[h * 64])[c];
        const float4 v4 = ((const float4*)&uv[256 + h * 64])[c];
        su[h] += s4.x * u4.x + s4.y * u4.y + s4.z * u4.z + s4.w * u4.w;
        sv[h] += o4.x * v4.x + o4.y * v4.y + o4.z * v4.z + o4.w * v4.w;
      }
    }

    // ---- stage 2: leaky-relu + softmax over m (pure intra-wave, lane == m) ----
#pragma unroll
    for (int h = 0; h < NHEAD; ++h) {
      float s = su[h] + sv[h];
      s = (s > 0.f) ? s : 0.2f * s;
      float mx = s;
#pragma unroll
      for (int d = 16; d >= 1; d >>= 1) mx = fmaxf(mx, __shfl_xor(mx, d, 32));
      const float e = __expf(s - mx);
      float sum = e;
#pragma unroll
      for (int d = 16; d >= 1; d >>= 1) sum += __shfl_xor(sum, d, 32);
      alp[(h * 16 + nloc) * 32 + lane] = e / sum;
    }
    __syncthreads();   // hot + alpha visible

    // ---- stage 3: g[h,n,:] = sum_m alpha * h_other row (lanes = k) ----
#pragma unroll
    for (int i = 0; i < 4; ++i) {
      const int q  = i * 8 + wave;       // 32 (h, n8) pairs this pass
      const int h  = q >> 3;
      const int n8 = q & 7;
      const float* hrow = &hot[(n8 * 32) * HOT_STRIDE];
      const float* arow = &alp[(h * 16 + p * 8 + n8) * 32];
      float a0 = 0.f, a1 = 0.f;
#pragma unroll 8
      for (int m = 0; m < 32; ++m) {
        const float al = arow[m];
        a0 += al * hrow[m * HOT_STRIDE + lane];
        a1 += al * hrow[m * HOT_STRIDE + 32 + lane];
      }
      float* grow = &g[(h * 16 + p * 8 + n8) * G_STRIDE];
      grow[lane]      = a0;
      grow[32 + lane] = a1;
    }
    __syncthreads();   // done with hot before next pass; g complete after p==1
  }

  // ---- stage 4: out[16x32] = relu(0.25 * sum_h g_h @ W_h) + bias via WMMA f32 ----
  // Two waves, one 16x16 tile each; head-sum folded into the C accumulator chain.
  if (wave < 2) {
    const int oc = wave;            // which 16-col half
    const int nn = lane & 15;
    const int hi = lane >> 4;       // K sub-pair select per ISA A/B layout
    v8f acc = {0.f, 0.f, 0.f, 0.f, 0.f, 0.f, 0.f, 0.f};
#pragma unroll
    for (int h = 0; h < NHEAD; ++h) {
      v2f Bf[16];
#pragma unroll
      for (int kc = 0; kc < 16; ++kc) {           // B frag: W[h][k][oc*16+nn]
        const int k = kc * 4 + hi * 2;
        const float* wp = w + (h * F_IN + k) * F_OUT + oc * 16 + nn;
        Bf[kc].x = wp[0];
        Bf[kc].y = wp[F_OUT];
      }
#pragma unroll
      for (int kc = 0; kc < 16; ++kc) {           // A frag from g in LDS
        const int k = kc * 4 + hi * 2;
        const v2f Af = *(const v2f*)&g[(h * 16 + nn) * G_STRIDE + k];
        acc = __builtin_amdgcn_wmma_f32_16x16x4_f32(false, Af, false, Bf[kc],
                                                    (short)0, acc, false, false);
      }
    }
    const float bb = bias[oc * 16 + nn];
#pragma unroll
    for (int j = 0; j < 8; ++j) {
      const int mrow = j + 8 * hi;               // C/D layout: vgpr j, lane-half hi
      float val = acc[j] * 0.25f;
      val = fmaxf(val, 0.f) + bb;
      out[(long)(n0 + mrow) * F_OUT + oc * 16 + nn] = val;
    }
  }
}

extern "C" void kernel_launch(void* const* d_in, const int* in_sizes, int n_in,
                              void* d_out, int out_size, void* d_ws, size_t ws_size,
                              hipStream_t stream) {
  const float* hs   = (const float*)d_in[0];   // h_self  [8000,32,64]
  const float* ho   = (const float*)d_in[1];   // h_other [8000,32,64]
  // d_in[2]=goal, d_in[3]=action: unused by the reference
  const float* w    = (const float*)d_in[4];   // [4,64,32]
  const float* a    = (const float*)d_in[5];   // [64]
  const float* bias = (const float*)d_in[6];   // [32]
  float* uv  = (float*)d_ws;                   // 512 floats: u[4][64], v[4][64]
  float* out = (float*)d_out;

  (void)in_sizes; (void)n_in; (void)out_size; (void)ws_size;

  hipFuncSetAttribute((const void*)gat_main,
                      hipFuncAttributeMaxDynamicSharedMemorySize, SM_BYTES);

  gat_prep<<<1, 256, 0, stream>>>(w, a, uv);
  gat_main<<<N_TOT / 16, 256, SM_BYTES, stream>>>(hs, ho, w, bias, uv, out);
}